// SimpleGCN_91139206021791
// MI455X (gfx1250) — compile-verified
//
#include <hip/hip_runtime.h>
#include <hip/hip_bf16.h>
#include <math.h>

typedef __attribute__((ext_vector_type(2))) float v2f;
typedef __attribute__((ext_vector_type(8))) float v8f;

#define N_GRAPHS 512

// ---------------------------------------------------------------- zero fill
__global__ void gcn_zero_kernel(float* __restrict__ p, long n) {
    long i = (long)blockIdx.x * blockDim.x + threadIdx.x;
    if (i < n) p[i] = 0.0f;
}

// ------------------------------------------- edge weights + structural degrees
__global__ void gcn_edge_prep_kernel(const float* __restrict__ dist,
                                     const int* __restrict__ src,
                                     const int* __restrict__ dst,
                                     float* __restrict__ ew,
                                     float* __restrict__ deg_out,
                                     float* __restrict__ deg_in, int E) {
    int e = blockIdx.x * blockDim.x + threadIdx.x;
    if (e >= E) return;
    float d = dist[e];
    ew[e] = expf(-(d * d) * (1.0f / 64.0f));   // EDGE_LENGTHSCALE^2 = 64
    atomicAdd(&deg_out[src[e]], 1.0f);
    atomicAdd(&deg_in[dst[e]], 1.0f);
}

// ----------------------- degree -> norm (in place) + per-graph node counts
__global__ void gcn_node_prep_kernel(float* __restrict__ deg_out,
                                     float* __restrict__ deg_in,
                                     const int* __restrict__ gids,
                                     float* __restrict__ counts, int N) {
    int n = blockIdx.x * blockDim.x + threadIdx.x;
    if (n >= N) return;
    deg_out[n] = 1.0f / sqrtf(fmaxf(deg_out[n], 1.0f));   // norm_src
    deg_in[n]  = 1.0f / sqrtf(fmaxf(deg_in[n],  1.0f));   // norm_dst
    atomicAdd(&counts[gids[n]], 1.0f);
}

// ------------- tiny GEMM: table_embed[95,64] = atom_table[95,200] @ W_embed + b
__global__ void gcn_embed_table_kernel(const float* __restrict__ atom_table,
                                       const float* __restrict__ W_embed,
                                       const float* __restrict__ b_embed,
                                       float* __restrict__ te) {
    int r = blockIdx.x;      // 95 rows
    int c = threadIdx.x;     // 64 cols
    float acc = b_embed[c];
    for (int k = 0; k < 200; ++k)
        acc = fmaf(atom_table[r * 200 + k], W_embed[k * 64 + c], acc);
    te[r * 64 + c] = acc;
}

// --------------- h[n] = table_embed[type[n]] * norm_src[n]  (wave per node)
__global__ void gcn_gather_scale_kernel(const int* __restrict__ types,
                                        const float* __restrict__ te,
                                        const float* __restrict__ norm_src,
                                        float* __restrict__ h, int N) {
    int idx  = blockIdx.x * blockDim.x + threadIdx.x;
    int n    = idx >> 5;
    int lane = idx & 31;
    if (n >= N) return;
    float s = norm_src[n];
    const float2 v = ((const float2*)(te + types[n] * 64))[lane];
    ((float2*)(h + (long)n * 64))[lane] = make_float2(v.x * s, v.y * s);
}

// ------------- scatter: agg[dst] += h[src] * ew   (wave per edge, 2 f32/lane)
__global__ void gcn_edge_scatter_kernel(const int* __restrict__ src,
                                        const int* __restrict__ dst,
                                        const float* __restrict__ ew,
                                        const float* __restrict__ h,
                                        float* __restrict__ agg, int E) {
    int e    = blockIdx.x * 8 + (threadIdx.x >> 5);
    int lane = threadIdx.x & 31;
    if (e >= E) return;
    int s = src[e], d = dst[e];
    float w = ew[e];
    const float2 v = *(const float2*)(h + (long)s * 64 + lane * 2);
    float* out = agg + (long)d * 64 + lane * 2;
    atomicAdd(out,     v.x * w);
    atomicAdd(out + 1, v.y * w);
}

// ---------------- WMMA GEMM [N,64]@[64,64] + fused GraphConv epilogue
// mode 0: out_h = relu(agg@W * norm_dst + b) * norm_src   (feeds next conv)
// mode 1: pooled[gid] += agg@W * norm_dst + b             (fused readout)
__global__ void gcn_gemm_conv_kernel(const float* __restrict__ agg,
                                     const float* __restrict__ W,
                                     const float* __restrict__ b,
                                     const float* __restrict__ norm_dst,
                                     const float* __restrict__ norm_src,
                                     float* __restrict__ out_h,
                                     const int* __restrict__ gids,
                                     float* __restrict__ pooled,
                                     int N, int numRowTiles, int mode) {
    const int wid = (blockIdx.x * blockDim.x + threadIdx.x) >> 5;
    if (wid >= numRowTiles * 4) return;          // wave-uniform: EXEC stays all-1
    const int lane    = threadIdx.x & 31;
    const int row0    = (wid >> 2) << 4;
    const int col0    = (wid & 3) << 4;
    const int m       = lane & 15;               // A row / B,C,D column
    const int g       = lane >> 4;               // lane group

    v8f acc = {0.f, 0.f, 0.f, 0.f, 0.f, 0.f, 0.f, 0.f};
    const float* arow = agg + (long)(row0 + m) * 64;
#pragma unroll
    for (int kb = 0; kb < 64; kb += 4) {
        // A 16x4 f32: vgpr j holds K = kb + 2g + j  (contiguous pair per lane)
        const float2 av = *(const float2*)(arow + kb + 2 * g);
        v2f a;  a.x = av.x;  a.y = av.y;
        // B 4x16 f32: vgpr j holds W[kb+2g+j][col0+m]
        v2f bm;
        bm.x = W[(kb + 2 * g + 0) * 64 + col0 + m];
        bm.y = W[(kb + 2 * g + 1) * 64 + col0 + m];
        acc = __builtin_amdgcn_wmma_f32_16x16x4_f32(false, a, false, bm,
                                                    (short)0, acc, false, false);
    }

    const float bias = b[col0 + m];
#pragma unroll
    for (int r = 0; r < 8; ++r) {
        const int row = row0 + r + 8 * g;        // C/D layout: M = r + 8g
        if (row < N) {
            float v = acc[r] * norm_dst[row] + bias;
            if (mode == 0) {
                v = fmaxf(v, 0.0f) * norm_src[row];
                out_h[(long)row * 64 + col0 + m] = v;
            } else {
                atomicAdd(&pooled[gids[row] * 64 + col0 + m], v);
            }
        }
    }
}

// ---------------------------------------------------- pooled /= count
__global__ void gcn_finalize_kernel(float* __restrict__ out,
                                    const float* __restrict__ counts) {
    int i = blockIdx.x * blockDim.x + threadIdx.x;   // 512*64
    out[i] = out[i] / fmaxf(counts[i >> 6], 1.0f);
}

extern "C" void kernel_launch(void* const* d_in, const int* in_sizes, int n_in,
                              void* d_out, int out_size, void* d_ws, size_t ws_size,
                              hipStream_t stream) {
    const int*   atom_types = (const int*)  d_in[0];
    const int*   src        = (const int*)  d_in[1];
    const int*   dst        = (const int*)  d_in[2];
    const float* distances  = (const float*)d_in[3];
    const int*   graph_ids  = (const int*)  d_in[4];
    const float* atom_table = (const float*)d_in[5];
    const float* W_embed    = (const float*)d_in[6];
    const float* b_embed    = (const float*)d_in[7];
    const float* W1         = (const float*)d_in[8];
    const float* b1         = (const float*)d_in[9];
    const float* W2         = (const float*)d_in[10];
    const float* b2         = (const float*)d_in[11];
    float* out = (float*)d_out;

    const int N = in_sizes[0];
    const int E = in_sizes[1];
    const int numRowTiles = (N + 15) / 16;
    const long Npad = (long)numRowTiles * 16;

    // workspace carve-out (floats)
    float* ws       = (float*)d_ws;
    float* deg_out  = ws;  ws += Npad;          // -> norm_src
    float* deg_in   = ws;  ws += Npad;          // -> norm_dst
    float* counts   = ws;  ws += N_GRAPHS;
    float* ew       = ws;  ws += E;
    float* te       = ws;  ws += 95 * 64;
    float* h        = ws;  ws += Npad * 64;
    float* agg      = ws;  ws += Npad * 64;

    const int B = 256;
    // zero accumulators (deg_out|deg_in|counts are contiguous)
    {
        long n0 = 2 * Npad + N_GRAPHS;
        gcn_zero_kernel<<<(unsigned)((n0 + B - 1) / B), B, 0, stream>>>(deg_out, n0);
        long n1 = Npad * 64;
        gcn_zero_kernel<<<(unsigned)((n1 + B - 1) / B), B, 0, stream>>>(agg, n1);
        long n2 = N_GRAPHS * 64;
        gcn_zero_kernel<<<(unsigned)((n2 + B - 1) / B), B, 0, stream>>>(out, n2);
    }

    gcn_edge_prep_kernel<<<(E + B - 1) / B, B, 0, stream>>>(distances, src, dst,
                                                            ew, deg_out, deg_in, E);
    gcn_node_prep_kernel<<<(N + B - 1) / B, B, 0, stream>>>(deg_out, deg_in,
                                                            graph_ids, counts, N);
    gcn_embed_table_kernel<<<95, 64, 0, stream>>>(atom_table, W_embed, b_embed, te);
    {
        long t = (long)N * 32;
        gcn_gather_scale_kernel<<<(unsigned)((t + B - 1) / B), B, 0, stream>>>(
            atom_types, te, deg_out, h, N);
    }

    const unsigned scatterGrid = (E + 7) / 8;                 // 8 edges/block
    const unsigned gemmGrid = (unsigned)(((long)numRowTiles * 4 * 32 + B - 1) / B);

    // ---- conv1: scatter then WMMA-GEMM (epilogue: relu * norm_dst + b1, pre-scale norm_src)
    gcn_edge_scatter_kernel<<<scatterGrid, B, 0, stream>>>(src, dst, ew, h, agg, E);
    gcn_gemm_conv_kernel<<<gemmGrid, B, 0, stream>>>(agg, W1, b1, deg_in, deg_out,
                                                     h, graph_ids, out,
                                                     N, numRowTiles, 0);
    // ---- conv2: re-zero agg, scatter, WMMA-GEMM with fused mean-pool accumulate
    {
        long n1 = Npad * 64;
        gcn_zero_kernel<<<(unsigned)((n1 + B - 1) / B), B, 0, stream>>>(agg, n1);
    }
    gcn_edge_scatter_kernel<<<scatterGrid, B, 0, stream>>>(src, dst, ew, h, agg, E);
    gcn_gemm_conv_kernel<<<gemmGrid, B, 0, stream>>>(agg, W2, b2, deg_in, deg_out,
                                                     h, graph_ids, out,
                                                     N, numRowTiles, 1);

    gcn_finalize_kernel<<<(N_GRAPHS * 64) / B, B, 0, stream>>>(out, counts);
}